// RoiAlign_68899865362766
// MI455X (gfx1250) — compile-verified
//
#include <hip/hip_runtime.h>
#include <stdint.h>

// ROI-Align (matches reference incl. its double application of SCALE = 1/16).
// fm: (1,256,50,50) f32, proposals: (1024,4) f32 [x1,y1,x2,y2]
// out: (1024,256,7,7) f32
//
// Data path (fast, taken for this input distribution):
//   1) per-proposal ROI tile (<=6x8 pixels, 64 channels/group) staged into LDS
//      with GLOBAL_LOAD_ASYNC_TO_LDS_B32 (ASYNCcnt), s_wait_asynccnt + barrier
//   2) bilinear samples built from LDS tile into bin-major X[c,p'] in LDS
//   3) per-bin 4-sample mean via V_WMMA_F32_16X16X4_F32, W block-diagonal 0.25
// Fallback (block-uniform, if the runtime bounding box exceeds the tile):
//   direct global gathers, identical math.

#define C_CH   256
#define FH     50
#define FW     50
#define OUTSZ  7
#define NBINS  49      // 7*7
#define NSAMP  196     // 49 bins * 4 samples, bin-major
#define XSTR   197     // LDS row stride (coprime with 64 banks)
#define TH     6       // tile rows budget
#define TW     8       // tile cols budget
#define TPLANE 49      // TH*TW + 1 pad (coprime with 64 banks)

typedef __attribute__((ext_vector_type(2))) float v2f;
typedef __attribute__((ext_vector_type(8))) float v8f;

__global__ __launch_bounds__(128)
void roi_align_wmma_kernel(const float* __restrict__ fm,
                           const float* __restrict__ props,
                           float* __restrict__ out) {
  __shared__ int   s_i0[2][14];   // [0]=y axis, [1]=x axis
  __shared__ int   s_i1[2][14];
  __shared__ float s_hi[2][14];   // 1 - frac
  __shared__ float s_lo[2][14];   // frac
  __shared__ int   s_vl[2][14];   // validity
  __shared__ float sX[4][16 * XSTR];   // per-wave X[c_local][p'] buffers
  __shared__ float sTile[64 * TPLANE]; // staged ROI tile, 64 channels/group

  const int prop = blockIdx.x;
  const int tid  = threadIdx.x;
  const int lane = tid & 31;
  const int wv   = tid >> 5;      // wave id 0..3

  // ---- per-proposal geometry (threads 0..27; axis tables into LDS) ----
  if (tid < 28) {
    const int axis = (tid < 14) ? 0 : 1;       // 0: y, 1: x
    const int g    = tid - axis * 14;
    const float sc = 1.0f / 16.0f;
    // column order in proposals: x1,y1,x2,y2 ; reference scales by sc TWICE
    const int c1 = axis ? 0 : 1;
    const int c2 = axis ? 2 : 3;
    float v1 = (props[prop * 4 + c1] * sc) * sc;
    float v2 = (props[prop * 4 + c2] * sc) * sc;
    float roi = fmaxf(v2 - v1, 1.0f);
    float bsz = roi / (float)OUTSZ;
    float grid = ((float)g + 0.5f) / 2.0f;     // sr = 2
    float v = v1 + bsz * grid;
    const float L = 50.0f;                     // H == W == 50
    int vld = (v >= -1.0f) && (v <= L);
    float vc = fminf(fmaxf(v, 0.0f), L - 1.0f);
    int i0 = (int)floorf(vc);
    if (i0 > FH - 1) i0 = FH - 1;
    int i1 = i0 + 1;
    if (i1 > FH - 1) i1 = FH - 1;
    float l = vc - (float)i0;
    s_i0[axis][g] = i0;
    s_i1[axis][g] = i1;
    s_hi[axis][g] = 1.0f - l;
    s_lo[axis][g] = l;
    s_vl[axis][g] = vld;
  }
  __syncthreads();

  // Runtime bounding box of all sample corners (axis tables are monotonic).
  const int ty0 = s_i0[0][0], ty1 = s_i1[0][13];
  const int tx0 = s_i0[1][0], tx1 = s_i1[1][13];
  const int th  = ty1 - ty0 + 1;
  const int tw  = tx1 - tx0 + 1;
  const bool fast = (th <= TH) && (tw <= TW);   // block-uniform

  const uint32_t tlds   = (uint32_t)(uintptr_t)(void*)sTile; // LDS byte offset
  const uint64_t fmbase = (uint64_t)(uintptr_t)fm;

  if (!fast) {
    // warm caches for the fallback path (gfx1250 global_prefetch)
    for (int c = tid; c < C_CH; c += 128) {
      __builtin_prefetch(fm + ((size_t)c * FH + ty0) * FW + tx0, 0, 1);
    }
  }

  const int n16 = lane & 15;
  const int h   = lane >> 4;

  for (int it = 0; it < 4; ++it) {
    const int ctile = it * 4 + wv;            // 16 channel-tiles over 4 waves
    float* Xw = sX[wv];

    if (fast) {
      // ---- Stage 64-channel ROI tile into LDS via async loads ----
      const int chbase = it * 64;
      for (int idx = tid; idx < 64 * TH * TW; idx += 128) {
        int c   = idx / (TH * TW);
        int rem = idx - c * (TH * TW);
        int r   = rem >> 3;       // TW == 8
        int cc  = rem & 7;
        if (r < th && cc < tw) {
          uint32_t ldst = tlds + 4u * (uint32_t)(c * TPLANE + r * TW + cc);
          uint32_t goff = 4u * (uint32_t)((chbase + c) * (FH * FW) +
                                          (ty0 + r) * FW + (tx0 + cc));
          asm volatile("global_load_async_to_lds_b32 %0, %1, %2"
                       :
                       : "v"(ldst), "v"(goff), "s"(fmbase)
                       : "memory");
        }
      }
      asm volatile("s_wait_asynccnt 0" ::: "memory");
      __syncthreads();

      // ---- Phase 1 (fast): bilinear from LDS tile into X, bin-major ----
      for (int i = 0; i < 7; ++i) {
        int p  = i * 32 + lane;
        int ok = p < NSAMP;
        int pp = ok ? p : NSAMP - 1;
        int bin = pp >> 2;
        int s   = pp & 3;
        int by = bin / 7;
        int bx = bin - by * 7;
        int gy = 2 * by + (s >> 1);
        int gx = 2 * bx + (s & 1);
        int iy0 = s_i0[0][gy], iy1 = s_i1[0][gy];
        int ix0 = s_i0[1][gx], ix1 = s_i1[1][gx];
        float hy = s_hi[0][gy], ly = s_lo[0][gy];
        float hx = s_hi[1][gx], lx = s_lo[1][gx];
        float w00 = hy * hx, w01 = hy * lx, w10 = ly * hx, w11 = ly * lx;
        int vld = s_vl[0][gy] & s_vl[1][gx];
        int t00 = (iy0 - ty0) * TW + (ix0 - tx0);
        int t01 = (iy0 - ty0) * TW + (ix1 - tx0);
        int t10 = (iy1 - ty0) * TW + (ix0 - tx0);
        int t11 = (iy1 - ty0) * TW + (ix1 - tx0);
        const float* tb = &sTile[(wv * 16) * TPLANE];
        #pragma unroll 4
        for (int cl = 0; cl < 16; ++cl) {
          const float* tp = tb + cl * TPLANE;
          float v = tp[t00] * w00 + tp[t01] * w01 + tp[t10] * w10 + tp[t11] * w11;
          v = vld ? v : 0.0f;
          if (ok) Xw[cl * XSTR + pp] = v;
        }
      }
    } else {
      // ---- Phase 1 (fallback): bilinear from global memory into X ----
      for (int i = 0; i < 7; ++i) {
        int p  = i * 32 + lane;
        int ok = p < NSAMP;
        int pp = ok ? p : NSAMP - 1;
        int bin = pp >> 2;
        int s   = pp & 3;
        int by = bin / 7;
        int bx = bin - by * 7;
        int gy = 2 * by + (s >> 1);
        int gx = 2 * bx + (s & 1);
        int iy0 = s_i0[0][gy], iy1 = s_i1[0][gy];
        int ix0 = s_i0[1][gx], ix1 = s_i1[1][gx];
        float hy = s_hi[0][gy], ly = s_lo[0][gy];
        float hx = s_hi[1][gx], lx = s_lo[1][gx];
        float w00 = hy * hx, w01 = hy * lx, w10 = ly * hx, w11 = ly * lx;
        int vld = s_vl[0][gy] & s_vl[1][gx];
        int o00 = iy0 * FW + ix0, o01 = iy0 * FW + ix1;
        int o10 = iy1 * FW + ix0, o11 = iy1 * FW + ix1;
        const float* base = fm + (size_t)(ctile * 16) * (FH * FW);
        #pragma unroll 4
        for (int cl = 0; cl < 16; ++cl) {
          const float* bp = base + (size_t)cl * (FH * FW);
          float v = bp[o00] * w00 + bp[o01] * w01 + bp[o10] * w10 + bp[o11] * w11;
          v = vld ? v : 0.0f;
          if (ok) Xw[cl * XSTR + pp] = v;
        }
      }
    }
    __syncthreads();

    // ---- Phase 2: WMMA reduction; each k-step lights output column kt ----
    for (int nt = 0; nt < 4; ++nt) {
      v8f acc = {};
      for (int kt = 0; kt < 16; ++kt) {
        int bk = nt * 16 + kt;                 // global bin index on K chain
        v2f a;
        if (bk < NBINS) {                      // uniform branch: EXEC untouched
          a[0] = Xw[n16 * XSTR + 4 * bk + 2 * h];
          a[1] = Xw[n16 * XSTR + 4 * bk + 2 * h + 1];
        } else {
          a[0] = 0.0f; a[1] = 0.0f;
        }
        float bw = ((n16 == kt) && (bk < NBINS)) ? 0.25f : 0.0f;
        v2f b; b[0] = bw; b[1] = bw;
        acc = __builtin_amdgcn_wmma_f32_16x16x4_f32(
            /*neg_a=*/false, a, /*neg_b=*/false, b,
            /*c_mod=*/(short)0, acc, /*reuse_a=*/false, /*reuse_b=*/false);
      }
      // D layout: lane -> n = lane%16; m = 8*(lane/16) + r
      int bin = nt * 16 + n16;
      if (bin < NBINS) {
        #pragma unroll
        for (int r = 0; r < 8; ++r) {
          int c = ctile * 16 + h * 8 + r;
          out[((size_t)prop * C_CH + c) * NBINS + bin] = acc[r];
        }
      }
    }
    __syncthreads();
  }
}

extern "C" void kernel_launch(void* const* d_in, const int* in_sizes, int n_in,
                              void* d_out, int out_size, void* d_ws, size_t ws_size,
                              hipStream_t stream) {
  const float* fm    = (const float*)d_in[0];   // (1,256,50,50) f32
  const float* props = (const float*)d_in[1];   // (1024,4) f32
  float* out = (float*)d_out;                   // (1024,256,7,7) f32
  const int nprop = in_sizes[1] / 4;
  dim3 grid(nprop), block(128);
  hipLaunchKernelGGL(roi_align_wmma_kernel, grid, block, 0, stream,
                     fm, props, out);
}